// OwnLSTM_18141941859056
// MI455X (gfx1250) — compile-verified
//
#include <hip/hip_runtime.h>
#include <hip/hip_bf16.h>

typedef __attribute__((ext_vector_type(2))) float v2f;
typedef __attribute__((ext_vector_type(8))) float v8f;

#define Bsz   32
#define Ssz   512
#define Hsz   512
#define Gsz   2048      // 4*H
#define Msz   16384     // B*S

// ---------------------------------------------------------------------------
// zero init for h/c state
__global__ void zero_kernel(float* __restrict__ p, int n) {
    int i = blockIdx.x * 256 + threadIdx.x;
    if (i < n) p[i] = 0.0f;
}

// ---------------------------------------------------------------------------
// Gather embedding rows into [S, B, E] contiguous layout.
// grid = 16384 blocks, block = 128 threads, one float4 per thread.
__global__ void gather_emb(const int* __restrict__ x,
                           const float* __restrict__ emb,
                           float* __restrict__ X0) {
    int row = blockIdx.x;            // row = s*32 + b
    int s = row >> 5;
    int b = row & 31;
    int tok = x[b * Ssz + s];
    const float4* src = (const float4*)(emb + (size_t)tok * Hsz);
    float4* dst = (float4*)(X0 + (size_t)row * Hsz);
    dst[threadIdx.x] = src[threadIdx.x];
}

// ---------------------------------------------------------------------------
// C = A[M,512] * W[2048,512]^T + (b1 + b2), f32 WMMA 16x16x4.
// Block: 256 threads (8 waves), tile 128(M) x 128(N), K staged via LDS.
// Wave layout: 4 (M) x 2 (N); each wave: 32 rows x 64 cols = 2x4 16x16 tiles.
__global__ void __launch_bounds__(256)
gemm_bias(const float* __restrict__ A,
          const float* __restrict__ W,
          const float* __restrict__ b1,
          const float* __restrict__ b2,
          float* __restrict__ Out) {
    __shared__ float As[128][36];   // stride 36 floats = 144B: 16B aligned, conflict-free
    __shared__ float Bs[128][36];

    const int tid  = threadIdx.x;
    const int wv   = tid >> 5;
    const int lane = tid & 31;
    const int lm   = lane & 15;
    const int kb   = (lane >> 4) << 1;     // 0 or 2
    const int wm   = (wv >> 1) * 32;       // wave M offset within block tile
    const int wn   = (wv & 1) * 64;        // wave N offset within block tile
    const int bm   = blockIdx.x * 128;
    const int bn   = blockIdx.y * 128;

    v8f acc[2][4] = {};

    for (int kc = 0; kc < 512; kc += 32) {
        #pragma unroll
        for (int i = 0; i < 4; ++i) {
            int idx = tid + i * 256;
            int row = idx >> 3;
            int c4  = (idx & 7) << 2;
            *(float4*)&As[row][c4] =
                *(const float4*)(A + (size_t)(bm + row) * 512 + kc + c4);
            *(float4*)&Bs[row][c4] =
                *(const float4*)(W + (size_t)(bn + row) * 512 + kc + c4);
        }
        __syncthreads();

        #pragma unroll
        for (int kk = 0; kk < 32; kk += 4) {
            v2f a[2], b[4];
            a[0] = *(const v2f*)&As[wm + lm][kk + kb];
            a[1] = *(const v2f*)&As[wm + 16 + lm][kk + kb];
            #pragma unroll
            for (int j = 0; j < 4; ++j)
                b[j] = *(const v2f*)&Bs[wn + j * 16 + lm][kk + kb];
            #pragma unroll
            for (int i = 0; i < 2; ++i)
                #pragma unroll
                for (int j = 0; j < 4; ++j)
                    acc[i][j] = __builtin_amdgcn_wmma_f32_16x16x4_f32(
                        false, a[i], false, b[j], (short)0, acc[i][j],
                        false, false);
        }
        __syncthreads();
    }

    const int rbase = (lane >> 4) * 8;
    #pragma unroll
    for (int i = 0; i < 2; ++i) {
        #pragma unroll
        for (int j = 0; j < 4; ++j) {
            int col = bn + wn + j * 16 + lm;
            float bias = b1[col] + b2[col];
            #pragma unroll
            for (int e = 0; e < 8; ++e) {
                int row = bm + wm + i * 16 + rbase + e;
                Out[(size_t)row * Gsz + col] = acc[i][j][e] + bias;
            }
        }
    }
}

// ---------------------------------------------------------------------------
// One LSTM timestep: gates = xg_t + h_in @ Whh^T ; cell update.
// grid = (2, 32): blockIdx.x -> 16-row batch tile, blockIdx.y -> 16 h-units.
// block = 128 threads (4 waves); wave w computes gate quadrant w (i/f/g/o).
__global__ void __launch_bounds__(128)
lstm_step(const float* __restrict__ xg,    // [B, 4H] slice for this timestep
          const float* __restrict__ Whh,   // [4H, H]
          const float* __restrict__ h_in,  // [B, H]
          float* __restrict__ h_out,       // [B, H]
          float* __restrict__ c_st,        // [B, H] (in-place)
          float* __restrict__ hs) {        // [B, H] slice of hs output
    __shared__ float tile[4][16][17];

    const int tid  = threadIdx.x;
    const int w    = tid >> 5;            // gate index 0..3
    const int lane = tid & 31;
    const int lm   = lane & 15;
    const int kb   = (lane >> 4) << 1;
    const int m0   = blockIdx.x * 16;     // batch offset
    const int h0   = blockIdx.y * 16;     // hidden-unit offset
    const int g0   = w * Hsz + h0;        // gate column offset

    const float* Ap = h_in + (size_t)(m0 + lm) * Hsz + kb;
    const float* Bp = Whh + (size_t)(g0 + lm) * Hsz + kb;

    v8f acc = {};
    #pragma unroll 8
    for (int k = 0; k < Hsz; k += 4) {
        v2f a = *(const v2f*)(Ap + k);
        v2f b = *(const v2f*)(Bp + k);
        acc = __builtin_amdgcn_wmma_f32_16x16x4_f32(
            false, a, false, b, (short)0, acc, false, false);
    }

    const int rbase = (lane >> 4) * 8;
    #pragma unroll
    for (int e = 0; e < 8; ++e) {
        int brow = rbase + e;  // local batch row 0..15
        tile[w][brow][lm] = acc[e] + xg[(size_t)(m0 + brow) * Gsz + g0 + lm];
    }
    __syncthreads();

    #pragma unroll
    for (int e = tid; e < 256; e += 128) {
        int br = e >> 4;
        int hc = e & 15;
        float ig = tile[0][br][hc];
        float fg = tile[1][br][hc];
        float gg = tile[2][br][hc];
        float og = tile[3][br][hc];
        float si = 1.0f / (1.0f + __expf(-ig));
        float sf = 1.0f / (1.0f + __expf(-fg));
        float sg = tanhf(gg);
        float so = 1.0f / (1.0f + __expf(-og));
        int idx = (m0 + br) * Hsz + h0 + hc;
        float cn = sf * c_st[idx] + si * sg;
        c_st[idx] = cn;
        float hn = so * tanhf(cn);
        h_out[idx] = hn;
        hs[idx] = hn;
    }
}

// ---------------------------------------------------------------------------
// Masked mean-pool over S (hs is [S, B, H]) + FC to 4 logits.
// grid = 32 (one block per batch row), block = 256.
__global__ void __launch_bounds__(256)
pool_fc(const float* __restrict__ hs,     // [S, B, H]
        const int* __restrict__ mask,     // [B, S]
        const float* __restrict__ fw,     // [4, H]
        const float* __restrict__ fb,     // [4]
        float* __restrict__ out) {        // [B, 4]
    __shared__ float pooled[Hsz];
    __shared__ float msum_s;

    int b = blockIdx.x;
    int tid = threadIdx.x;

    if (tid == 0) {
        float ms = 0.0f;
        for (int s = 0; s < Ssz; ++s) ms += (float)mask[b * Ssz + s];
        msum_s = ms < 1e-9f ? 1e-9f : ms;
    }

    float acc0 = 0.0f, acc1 = 0.0f;
    for (int s = 0; s < Ssz; ++s) {
        float m = (float)mask[b * Ssz + s];
        const float* p = hs + (size_t)s * (Bsz * Hsz) + (size_t)b * Hsz;
        acc0 += p[tid] * m;
        acc1 += p[tid + 256] * m;
    }
    __syncthreads();
    float inv = 1.0f / msum_s;
    pooled[tid] = acc0 * inv;
    pooled[tid + 256] = acc1 * inv;
    __syncthreads();

    if (tid < 4) {
        float r = fb[tid];
        for (int h = 0; h < Hsz; ++h) r += pooled[h] * fw[tid * Hsz + h];
        out[b * 4 + tid] = r;
    }
}

// ---------------------------------------------------------------------------
extern "C" void kernel_launch(void* const* d_in, const int* in_sizes, int n_in,
                              void* d_out, int out_size, void* d_ws, size_t ws_size,
                              hipStream_t stream) {
    const int*   x    = (const int*)d_in[0];
    const int*   am   = (const int*)d_in[1];
    const float* emb  = (const float*)d_in[2];
    const float* Wih0 = (const float*)d_in[3];
    const float* Whh0 = (const float*)d_in[4];
    const float* bih0 = (const float*)d_in[5];
    const float* bhh0 = (const float*)d_in[6];
    const float* Wih1 = (const float*)d_in[7];
    const float* Whh1 = (const float*)d_in[8];
    const float* bih1 = (const float*)d_in[9];
    const float* bhh1 = (const float*)d_in[10];
    const float* fcw  = (const float*)d_in[11];
    const float* fcb  = (const float*)d_in[12];
    float* out = (float*)d_out;

    float* ws  = (float*)d_ws;
    float* X0  = ws;                                   // [M, 512]  (later reused as hs1)
    float* XG  = X0 + (size_t)Msz * Hsz;               // [M, 2048] (both layers)
    float* HS0 = XG + (size_t)Msz * Gsz;               // [M, 512]
    float* HB  = HS0 + (size_t)Msz * Hsz;              // 2 x [B, H] h double-buffer
    float* CS  = HB + 2 * Bsz * Hsz;                   // [B, H] c state

    const int stateN = 3 * Bsz * Hsz;                  // HB (both) + CS, contiguous

    // ---- layer 0 ----
    zero_kernel<<<(stateN + 255) / 256, 256, 0, stream>>>(HB, stateN);
    gather_emb<<<Msz, 128, 0, stream>>>(x, emb, X0);
    gemm_bias<<<dim3(Msz / 128, Gsz / 128), 256, 0, stream>>>(X0, Wih0, bih0, bhh0, XG);
    for (int s = 0; s < Ssz; ++s) {
        lstm_step<<<dim3(2, 32), 128, 0, stream>>>(
            XG + (size_t)s * Bsz * Gsz, Whh0,
            HB + (s & 1) * (Bsz * Hsz), HB + ((s + 1) & 1) * (Bsz * Hsz),
            CS, HS0 + (size_t)s * Bsz * Hsz);
    }

    // ---- layer 1 ----
    zero_kernel<<<(stateN + 255) / 256, 256, 0, stream>>>(HB, stateN);
    gemm_bias<<<dim3(Msz / 128, Gsz / 128), 256, 0, stream>>>(HS0, Wih1, bih1, bhh1, XG);
    for (int s = 0; s < Ssz; ++s) {
        lstm_step<<<dim3(2, 32), 128, 0, stream>>>(
            XG + (size_t)s * Bsz * Gsz, Whh1,
            HB + (s & 1) * (Bsz * Hsz), HB + ((s + 1) & 1) * (Bsz * Hsz),
            CS, X0 + (size_t)s * Bsz * Hsz);           // hs1 into X0 buffer
    }

    // ---- pool + fc ----
    pool_fc<<<Bsz, 256, 0, stream>>>(X0, am, fcw, fcb, out);
}